// BatchOnlineNorm_12429635354902
// MI455X (gfx1250) — compile-verified
//
#include <hip/hip_runtime.h>
#include <hip/hip_bf16.h>

typedef __attribute__((ext_vector_type(2))) float v2f;
typedef __attribute__((ext_vector_type(8))) float v8f;

#define AFWD 0.999f
#define EPSV 1e-5f
#define CCH  256            // channels
#define HW   4096           // 64*64 spatial
#define NB   32             // batch
#define NCHUNK 16           // spatial chunks per sample in K1
#define POS_PER_CHUNK (HW / NCHUNK)   // 256

// ---------------------------------------------------------------------------
// K1: per-(b, c) partial spatial moments via WMMA ones-matrix reduction.
//   P1[b][chunk][c] = sum over chunk's 256 positions of x
//   P2[b][chunk][c] = sum of x^2
// Each wave handles two 16-channel tiles. Per inner step, the wave's B matrix
// holds 4 spatial positions x 16 channels (lane<16: k=0,1 ; lane>=16: k=2,3),
// and D = ones(16x4) * B + C accumulates column sums in every output row.
// ---------------------------------------------------------------------------
__global__ __launch_bounds__(256) void k1_stats(const float* __restrict__ x,
                                                float* __restrict__ P1,
                                                float* __restrict__ P2) {
    const int chunk = blockIdx.x;          // 0..15
    const int b     = blockIdx.y;          // 0..31
    const int wave  = threadIdx.x >> 5;    // 0..7
    const int lane  = threadIdx.x & 31;
    const int half  = lane >> 4;           // 0/1 -> owns k={0,1} or k={2,3}
    const int li    = lane & 15;           // column (channel) within tile

    const int p0 = chunk * POS_PER_CHUNK;

    v2f ones;
    ones[0] = 1.0f;
    ones[1] = 1.0f;

#pragma unroll
    for (int t = 0; t < 2; ++t) {
        const int c0 = (wave + t * 8) * 16;              // channel tile base
        const float* base = x + ((size_t)b * HW) * CCH + c0 + li;

        v8f s1 = {};
        v8f s2 = {};

        for (int g = 0; g < POS_PER_CHUNK / 4; ++g) {
            const int p = p0 + g * 4 + half * 2;
            float e0 = base[(size_t)p * CCH];            // B row k (this half)
            float e1 = base[(size_t)(p + 1) * CCH];      // B row k+1
            v2f bm, bq;
            bm[0] = e0;       bm[1] = e1;
            bq[0] = e0 * e0;  bq[1] = e1 * e1;
            s1 = __builtin_amdgcn_wmma_f32_16x16x4_f32(
                     false, ones, false, bm, (short)0, s1, false, false);
            s2 = __builtin_amdgcn_wmma_f32_16x16x4_f32(
                     false, ones, false, bq, (short)0, s2, false, false);
        }
        // Row M=0 of D (lanes 0..15, VGPR0) holds the per-channel sums.
        if (half == 0) {
            size_t o = (((size_t)b * NCHUNK) + chunk) * CCH + c0 + li;
            P1[o] = s1[0];
            P2[o] = s2[0];
        }
    }
}

// ---------------------------------------------------------------------------
// K2: one block, one thread per channel. Runs the B=32 streaming recurrence,
// computes the per-sample RMS in closed form, and folds normalization +
// gamma/beta + layer scaling into one affine map per (b, c):
//   final_out = Acoef[b][c] * x + Bcoef[b][c]
// ---------------------------------------------------------------------------
__global__ __launch_bounds__(256) void k2_recur(const float* __restrict__ P1,
                                                const float* __restrict__ P2,
                                                const float* __restrict__ gamma,
                                                const float* __restrict__ beta,
                                                const float* __restrict__ stream_mu,
                                                const float* __restrict__ stream_var,
                                                float* __restrict__ Acoef,
                                                float* __restrict__ Bcoef) {
    const int c = threadIdx.x;             // 0..255
    __shared__ float red[CCH];

    const float Nf = (float)HW;
    float mu  = stream_mu[c];
    float var = stream_var[c];
    const float g  = gamma[c];
    const float be = beta[c];

    for (int b = 0; b < NB; ++b) {
        float s1 = 0.0f, s2 = 0.0f;
        for (int ch = 0; ch < NCHUNK; ++ch) {
            size_t o = (((size_t)b * NCHUNK) + ch) * CCH + c;
            s1 += P1[o];
            s2 += P2[o];
        }
        const float rinv   = rsqrtf(var + EPSV);
        const float sum_d  = s1 - Nf * mu;                         // sum(x - mu)
        const float sum_d2 = s2 - 2.0f * mu * s1 + Nf * mu * mu;   // sum((x-mu)^2)
        const float gr     = g * rinv;

        // per-channel contribution to sum_{h,w,c} out^2  (out = gr*(x-mu)+be)
        float contrib = gr * gr * sum_d2 + 2.0f * gr * be * sum_d + Nf * be * be;

        red[c] = contrib;
        __syncthreads();
        for (int s = CCH / 2; s > 0; s >>= 1) {
            if (c < s) red[c] += red[c + s];
            __syncthreads();
        }
        const float total = red[0];
        __syncthreads();                         // before red[] reuse next iter

        const float ms     = total / (Nf * (float)CCH);
        const float rmsinv = rsqrtf(ms + EPSV);

        Acoef[b * CCH + c] = gr * rmsinv;
        Bcoef[b * CCH + c] = (be - gr * mu) * rmsinv;

        // streaming update (uses incoming mu/var)
        var = AFWD * var + AFWD * (1.0f - AFWD) * (sum_d2 / Nf);
        mu  = AFWD * mu + (1.0f - AFWD) * (s1 / Nf);
    }
}

// ---------------------------------------------------------------------------
// K3: fused affine apply, float4 (b128) loads/stores. x is L2-resident after
// K1 (134 MB < 192 MB L2), so this pass's read mostly hits L2; the store is
// the second and final HBM traversal.
// ---------------------------------------------------------------------------
__global__ __launch_bounds__(256) void k3_apply(const float* __restrict__ x,
                                                const float* __restrict__ Acoef,
                                                const float* __restrict__ Bcoef,
                                                float* __restrict__ out) {
    const size_t idx = (size_t)blockIdx.x * blockDim.x + threadIdx.x; // float4 idx
    const int b  = (int)(idx >> 18);     // HW*C/4 = 262144 = 2^18 float4/sample
    const int c4 = (int)(idx & 63);      // 64 float4 channel groups per position

    const float4* xv = (const float4*)x;
    const float4* Av = (const float4*)Acoef;
    const float4* Bv = (const float4*)Bcoef;

    float4 xx = xv[idx];
    float4 A  = Av[b * 64 + c4];
    float4 B  = Bv[b * 64 + c4];

    float4 o;
    o.x = fmaf(A.x, xx.x, B.x);
    o.y = fmaf(A.y, xx.y, B.y);
    o.z = fmaf(A.z, xx.z, B.z);
    o.w = fmaf(A.w, xx.w, B.w);
    ((float4*)out)[idx] = o;
}

// ---------------------------------------------------------------------------
extern "C" void kernel_launch(void* const* d_in, const int* in_sizes, int n_in,
                              void* d_out, int out_size, void* d_ws, size_t ws_size,
                              hipStream_t stream) {
    const float* x          = (const float*)d_in[0];
    const float* gamma      = (const float*)d_in[1];
    const float* beta       = (const float*)d_in[2];
    const float* stream_mu  = (const float*)d_in[3];
    const float* stream_var = (const float*)d_in[4];
    float* out = (float*)d_out;

    // Workspace layout (floats): P1 | P2 | Acoef | Bcoef
    float* P1 = (float*)d_ws;                          // 32*16*256
    float* P2 = P1 + (size_t)NB * NCHUNK * CCH;        // 32*16*256
    float* Ac = P2 + (size_t)NB * NCHUNK * CCH;        // 32*256
    float* Bc = Ac + (size_t)NB * CCH;                 // 32*256

    dim3 g1(NCHUNK, NB);
    k1_stats<<<g1, 256, 0, stream>>>(x, P1, P2);
    k2_recur<<<1, CCH, 0, stream>>>(P1, P2, gamma, beta, stream_mu, stream_var, Ac, Bc);

    const size_t total_f4 = (size_t)NB * HW * CCH / 4;     // 8,388,608
    k3_apply<<<(int)(total_f4 / 256), 256, 0, stream>>>(x, Ac, Bc, out);
}